// GELU275_23648089932070
// MI455X (gfx1250) — compile-verified
//
#include <hip/hip_runtime.h>
#include <cstdint>
#include <cstddef>

#define AS1 __attribute__((address_space(1)))
#define AS3 __attribute__((address_space(3)))

typedef int   v4i __attribute__((vector_size(16)));
typedef float v4f __attribute__((ext_vector_type(4)));

constexpr int kB       = 8;
constexpr int kD       = 2048;                 // floats per row
constexpr int kThreads = 256;                  // 8 wave32
constexpr int kWaves   = kThreads / 32;
constexpr int kVec     = kD / 4;               // 512 float4 per row
constexpr int kIter    = kVec / 32;            // 16 float4 per lane per row

__device__ __forceinline__ float gelu_f(float x) {
    const float c0 = 0.7978845608028654f;      // sqrt(2/pi)
    const float c1 = 0.044715f;
    float x2 = x * x;
#if __has_builtin(__builtin_amdgcn_tanhf)
    // y = 0.5*x*(1 + tanh(c0*(x + c1*x^3))) with hardware v_tanh_f32
    float u  = x * fmaf(c0 * c1, x2, c0);
    float t  = __builtin_amdgcn_tanhf(u);
    float hx = 0.5f * x;
    return fmaf(hx, t, hx);
#else
    // 0.5*(1+tanh(u)) == sigmoid(2u) = 1/(1+exp(-2u))
    float u2 = x * fmaf(-2.0f * c0 * c1, x2, -2.0f * c0);
    float e  = __expf(u2);
    return x * __builtin_amdgcn_rcpf(1.0f + e);
#endif
}

// One block per t. Wave w handles batch row b=w. Single pass over x/out.
__global__ __launch_bounds__(kThreads) void gelu275_fused(
    const float* __restrict__ x,          // [B,T,D]
    const float* __restrict__ log_k_pos,  // [1]
    const float* __restrict__ pos_buf,    // [T,D]
    const float* __restrict__ pos_facil,  // [T]
    float* __restrict__ out,              // [B,T,D]
    int T)
{
    extern __shared__ __align__(16) float smem[];
    float* ybuf = smem;                 // [kB*kD]  x staged, then y in place
    float* pbuf = smem + kB * kD;       // [kD]
    float* red  = pbuf + kD;            // [3*kWaves]

    const int t    = blockIdx.x;
    const int wave = threadIdx.x >> 5;
    const int lane = threadIdx.x & 31;

    const v4f* __restrict__ gx = (const v4f*)(x + ((size_t)wave * T + t) * kD);
    const v4f* __restrict__ gp = (const v4f*)(pos_buf + (size_t)t * kD);
    v4f* __restrict__ go       = (v4f*)(out + ((size_t)wave * T + t) * kD);

    v4f* ly4 = (v4f*)(ybuf + wave * kD);
    v4f* lp4 = (v4f*)pbuf;

    // ---- Stage tile into LDS (async DMA path on gfx1250, tracked by ASYNCcnt) ----
#if __has_builtin(__builtin_amdgcn_global_load_async_to_lds_b128)
    {
        AS1 char* gx1 = (AS1 char*)(uintptr_t)gx;
        AS3 char* ly3 = (AS3 char*)(uint32_t)(uintptr_t)ly4;   // low 32 bits of generic shared addr = LDS offset
        #pragma unroll
        for (int i = 0; i < kIter; ++i) {
            int j = i * 32 + lane;
            __builtin_amdgcn_global_load_async_to_lds_b128(
                (AS1 v4i*)(gx1 + (size_t)j * 16),
                (AS3 v4i*)(ly3 + j * 16), 0, 0);
        }
        AS1 char* gp1 = (AS1 char*)(uintptr_t)gp;
        AS3 char* lp3 = (AS3 char*)(uint32_t)(uintptr_t)lp4;
        #pragma unroll
        for (int i = 0; i < 2; ++i) {                          // each wave stages 2/16 chunks of p
            int j = (wave * 2 + i) * 32 + lane;
            __builtin_amdgcn_global_load_async_to_lds_b128(
                (AS1 v4i*)(gp1 + (size_t)j * 16),
                (AS3 v4i*)(lp3 + j * 16), 0, 0);
        }
        asm volatile("s_wait_asynccnt 0" ::: "memory");
    }
#else
    #pragma unroll
    for (int i = 0; i < kIter; ++i) { int j = i * 32 + lane; ly4[j] = gx[j]; }
    #pragma unroll
    for (int i = 0; i < 2; ++i)     { int j = (wave * 2 + i) * 32 + lane; lp4[j] = gp[j]; }
#endif
    __syncthreads();

    // ---- gelu + partial reductions (dot, ||y||^2), y written back in place ----
    float dot = 0.f, ss = 0.f, ssp = 0.f;
    #pragma unroll
    for (int i = 0; i < kIter; ++i) {
        int j = i * 32 + lane;
        v4f xv = ly4[j];
        v4f pv = lp4[j];
        v4f yv;
        yv.x = gelu_f(xv.x); yv.y = gelu_f(xv.y);
        yv.z = gelu_f(xv.z); yv.w = gelu_f(xv.w);
        ly4[j] = yv;
        dot = fmaf(yv.x, pv.x, fmaf(yv.y, pv.y, fmaf(yv.z, pv.z, fmaf(yv.w, pv.w, dot))));
        ss  = fmaf(yv.x, yv.x, fmaf(yv.y, yv.y, fmaf(yv.z, yv.z, fmaf(yv.w, yv.w, ss))));
    }
    // ||p||^2 is shared across all b: compute it once (wave 0 only)
    if (wave == 0) {
        #pragma unroll
        for (int i = 0; i < kIter; ++i) {
            v4f pv = lp4[i * 32 + lane];
            ssp = fmaf(pv.x, pv.x, fmaf(pv.y, pv.y, fmaf(pv.z, pv.z, fmaf(pv.w, pv.w, ssp))));
        }
    }
    #pragma unroll
    for (int m = 16; m >= 1; m >>= 1) {
        dot += __shfl_xor(dot, m, 32);
        ss  += __shfl_xor(ss,  m, 32);
        ssp += __shfl_xor(ssp, m, 32);
    }
    if (lane == 0) {
        red[wave * 3 + 0] = dot;
        red[wave * 3 + 1] = ss;
        red[wave * 3 + 2] = ssp;   // only red[2] (wave 0) is consumed
    }
    __syncthreads();

    // ---- cross-batch fire mask + per-b gate ----
    float kp = expf(log_k_pos[0]);
    kp = fminf(fmaxf(kp, 0.01f), 5.0f);
    float np = fmaxf(sqrtf(red[2]), 1e-12f);   // ||p||
    float simsum = 0.f, my_sim = 0.f;
    #pragma unroll
    for (int b = 0; b < kB; ++b) {
        float nb = fmaxf(sqrtf(red[b * 3 + 1]), 1e-12f);
        float s  = red[b * 3 + 0] / (nb * np);
        simsum += s;
        if (b == wave) my_sim = s;
    }
    float pf    = pos_facil[t];
    float facil = (simsum * 0.125f > 0.85f) ? fminf(pf * 2.0f, 16.0f) : pf;
    float gate  = fminf(1.0f + kp * (facil - 1.0f) * my_sim, 8.0f);

    // ---- stream y * gate to global (b128, coalesced, nontemporal) ----
    #pragma unroll
    for (int i = 0; i < kIter; ++i) {
        int j = i * 32 + lane;
        v4f ov = ly4[j] * gate;
        __builtin_nontemporal_store(ov, &go[j]);
    }
}

extern "C" void kernel_launch(void* const* d_in, const int* in_sizes, int n_in,
                              void* d_out, int out_size, void* d_ws, size_t ws_size,
                              hipStream_t stream) {
    (void)n_in; (void)out_size; (void)d_ws; (void)ws_size;
    const float* x   = (const float*)d_in[0];
    const float* lk  = (const float*)d_in[1];
    const float* pb  = (const float*)d_in[2];
    const float* pf  = (const float*)d_in[3];
    float* out       = (float*)d_out;

    const int T = in_sizes[3];  // pos_facil length (4096)
    const size_t shmem = (size_t)(kB * kD + kD + 3 * kWaves) * sizeof(float);
    gelu275_fused<<<dim3(T), dim3(kThreads), shmem, stream>>>(x, lk, pb, pf, out, T);
}